// NodeGeneratorTopology_67559835566822
// MI455X (gfx1250) — compile-verified
//
#include <hip/hip_runtime.h>
#include <math.h>

// ---------------------------------------------------------------------------
// GCN forward (2x GCNConv + linear + sigmoid) for MI455X / gfx1250.
// N = 50000 nodes, E = 800000 edges, all feature dims = 128, fp32.
//
// GEMMs use V_WMMA_F32_16X16X4_F32 (full fp32 matrix path, matches reference
// precision; GEMM cost is negligible vs. the edge scatter, which is the
// 23.3 TB/s HBM-bound part of this workload).
// ---------------------------------------------------------------------------

#define RRELU_SLOPE 0.2291666666666667f /* (1/8 + 1/3)/2 = 11/48 */
#define FEAT 128

typedef __attribute__((ext_vector_type(2))) float v2f;
typedef __attribute__((ext_vector_type(8))) float v8f;

// ---------------- degree / normalization ----------------

__global__ void k_deg_init(float* __restrict__ deg, int n) {
    int i = blockIdx.x * blockDim.x + threadIdx.x;
    if (i < n) deg[i] = 1.0f;  // +1 self loop
}

__global__ void k_deg_count(const int* __restrict__ dst, float* __restrict__ deg, int e) {
    int i = blockIdx.x * blockDim.x + threadIdx.x;
    if (i < e) atomicAdd(&deg[dst[i]], 1.0f);
}

__global__ void k_dinv(float* __restrict__ deg, int n) {
    int i = blockIdx.x * blockDim.x + threadIdx.x;
    if (i < n) deg[i] = rsqrtf(deg[i]);  // in place: deg -> deg^-1/2
}

// ---------------- WMMA fp32 GEMM: H = act(X @ W + bias) ----------------
// X: [M,128], W: [128,128], H: [M,128].
// Block = 256 threads = 8 waves. Block handles a 16-row stripe; wave w
// handles column tile [16w, 16w+16). K-loop of 32 x V_WMMA_F32_16X16X4_F32.
//
// VGPR layouts per CDNA5 ISA 7.12.2 (wave32):
//   A 16x4 : lane m=lane%16 is row M=m; VGPR{0,1} hold K = 2*(lane/16)+{0,1}
//   B 4x16 : lane n=lane%16 is col N=n; VGPR{0,1} hold K = 2*(lane/16)+{0,1}
//   C/D    : VGPR v -> row M = v + 8*(lane/16), col N = lane%16

__global__ __launch_bounds__(256) void k_gemm_wmma(
    const float* __restrict__ X, const float* __restrict__ W,
    const float* __restrict__ bias,  // may be nullptr
    float* __restrict__ H, int M, int act /*0=none,1=sigmoid*/) {

    const int row0 = blockIdx.x << 4;
    const int wave = threadIdx.x >> 5;
    const int lane = threadIdx.x & 31;
    const int m    = lane & 15;
    const int kh   = lane >> 4;      // 0 or 1: which K-half this lane holds
    const int n0   = wave << 4;

    int arow = row0 + m;
    if (arow >= M) arow = M - 1;     // clamp (M=50000 is a multiple of 16 anyway)

    const float* __restrict__ ap = X + (size_t)arow * FEAT + 2 * kh;
    const float* __restrict__ bp = W + (size_t)(2 * kh) * FEAT + n0 + m;

    v8f c = {};
#pragma unroll 4
    for (int k = 0; k < FEAT; k += 4) {
        v2f a, b;
        a.x = ap[k];
        a.y = ap[k + 1];
        b.x = bp[(size_t)k * FEAT];
        b.y = bp[(size_t)(k + 1) * FEAT];
        c = __builtin_amdgcn_wmma_f32_16x16x4_f32(
            /*neg_a=*/false, a, /*neg_b=*/false, b,
            /*c_mod=*/(short)0, c, /*reuse_a=*/false, /*reuse_b=*/false);
    }

    const float bv = bias ? bias[n0 + m] : 0.0f;
#pragma unroll
    for (int v = 0; v < 8; ++v) {
        const int row = row0 + v + (kh << 3);
        if (row < M) {
            float val = c[v] + bv;
            if (act == 1) {
                // sigmoid via fast reciprocal (single v_rcp_f32, ~1 ulp):
                // avoids the IEEE div-scale/Newton sequence in the epilogue.
                val = __builtin_amdgcn_rcpf(1.0f + __expf(-val));
            }
            H[(size_t)row * FEAT + n0 + m] = val;
        }
    }
}

// ---------------- agg init: agg = h * dinv^2 + bias ----------------
// One thread per (node, 4 features); float4 vectorized.

__global__ void k_init_agg(const float* __restrict__ h, const float* __restrict__ dinv,
                           const float* __restrict__ bias, float* __restrict__ agg, int n) {
    int gid = blockIdx.x * blockDim.x + threadIdx.x;
    if (gid >= (n << 5)) return;
    const int node = gid >> 5;
    const int q    = (gid & 31) << 2;
    const float di = dinv[node];
    const float s  = di * di;
    const size_t base = (size_t)node * FEAT + q;
    const float4 hv = *(const float4*)(h + base);
    const float4 bv = *(const float4*)(bias + q);
    float4 o;
    o.x = hv.x * s + bv.x;
    o.y = hv.y * s + bv.y;
    o.z = hv.z * s + bv.z;
    o.w = hv.w * s + bv.w;
    *(float4*)(agg + base) = o;
}

// ---------------- edge scatter: agg[dst] += h[src] * dinv[src]*dinv[dst] ----
// One wave of 32 lanes covers one edge's 128 features (float4 per lane):
// fully coalesced 512B gather per edge, 4 f32 atomics per lane into L2.

__global__ void k_scatter(const float* __restrict__ h, const float* __restrict__ dinv,
                          const int* __restrict__ src, const int* __restrict__ dst,
                          float* __restrict__ agg, int e) {
    int gid = blockIdx.x * blockDim.x + threadIdx.x;
    if (gid >= (e << 5)) return;
    const int edge = gid >> 5;
    const int q    = (gid & 31) << 2;
    const int s    = src[edge];
    const int d    = dst[edge];
    const float coef = dinv[s] * dinv[d];
    const float4 hv = *(const float4*)(h + (size_t)s * FEAT + q);
    float* __restrict__ ap = agg + (size_t)d * FEAT + q;
    atomicAdd(ap + 0, hv.x * coef);
    atomicAdd(ap + 1, hv.y * coef);
    atomicAdd(ap + 2, hv.z * coef);
    atomicAdd(ap + 3, hv.w * coef);
}

// ---------------- rrelu (eval mode leaky relu), float4 in place ------------

__global__ void k_rrelu4(float* __restrict__ a, int total4) {
    int i = blockIdx.x * blockDim.x + threadIdx.x;
    if (i >= total4) return;
    float4 v = *(float4*)(a + (size_t)i * 4);
    v.x = v.x >= 0.0f ? v.x : v.x * RRELU_SLOPE;
    v.y = v.y >= 0.0f ? v.y : v.y * RRELU_SLOPE;
    v.z = v.z >= 0.0f ? v.z : v.z * RRELU_SLOPE;
    v.w = v.w >= 0.0f ? v.w : v.w * RRELU_SLOPE;
    *(float4*)(a + (size_t)i * 4) = v;
}

// ---------------------------------------------------------------------------

extern "C" void kernel_launch(void* const* d_in, const int* in_sizes, int n_in,
                              void* d_out, int out_size, void* d_ws, size_t ws_size,
                              hipStream_t stream) {
    const float* x  = (const float*)d_in[0];
    const int*   ei = (const int*)d_in[1];
    const float* W1 = (const float*)d_in[2];
    const float* b1 = (const float*)d_in[3];
    const float* W2 = (const float*)d_in[4];
    const float* b2 = (const float*)d_in[5];
    const float* Wl = (const float*)d_in[6];
    const float* bl = (const float*)d_in[7];
    float* out = (float*)d_out;

    const int N = in_sizes[0] / FEAT;  // 50000
    const int E = in_sizes[1] / 2;     // 800000
    const int* src = ei;
    const int* dst = ei + E;

    // Workspace layout: dinv[N] | H[N*128] | A[N*128]   (~51.4 MB)
    char* ws = (char*)d_ws;
    size_t off = 0;
    auto take = [&](size_t bytes) {
        void* p = ws + off;
        off += (bytes + 255) & ~(size_t)255;
        return p;
    };
    float* dinv = (float*)take((size_t)N * sizeof(float));
    float* H    = (float*)take((size_t)N * FEAT * sizeof(float));
    float* A    = (float*)take((size_t)N * FEAT * sizeof(float));

    const int TB = 256;
    const int gN  = (N + TB - 1) / TB;
    const int gE  = (E + TB - 1) / TB;
    const int nf4 = N << 5;            // N*32 float4-threads
    const int ef4 = E << 5;            // E*32 float4-threads
    const int gNf = (nf4 + TB - 1) / TB;
    const int gEf = (ef4 + TB - 1) / TB;
    const int gGemm = (N + 15) / 16;   // 3125 row tiles

    // ---- normalization coefficients (shared by both conv layers) ----
    k_deg_init<<<gN, TB, 0, stream>>>(dinv, N);
    k_deg_count<<<gE, TB, 0, stream>>>(dst, dinv, E);
    k_dinv<<<gN, TB, 0, stream>>>(dinv, N);

    // ---- layer 1: A = rrelu( norm_agg(x @ W1) + b1 ) ----
    k_gemm_wmma<<<gGemm, 256, 0, stream>>>(x, W1, nullptr, H, N, 0);
    k_init_agg<<<gNf, TB, 0, stream>>>(H, dinv, b1, A, N);
    k_scatter<<<gEf, TB, 0, stream>>>(H, dinv, src, dst, A, E);
    k_rrelu4<<<gNf, TB, 0, stream>>>(A, nf4);

    // ---- layer 2: A = rrelu( norm_agg(A @ W2) + b2 ) ----
    k_gemm_wmma<<<gGemm, 256, 0, stream>>>(A, W2, nullptr, H, N, 0);
    k_init_agg<<<gNf, TB, 0, stream>>>(H, dinv, b2, A, N);   // A free after GEMM
    k_scatter<<<gEf, TB, 0, stream>>>(H, dinv, src, dst, A, E);
    k_rrelu4<<<gNf, TB, 0, stream>>>(A, nf4);

    // ---- output layer: out = sigmoid(A @ Wl + bl) ----
    k_gemm_wmma<<<gGemm, 256, 0, stream>>>(A, Wl, bl, out, N, 1);
}